// Fingerprint_48103633715874
// MI455X (gfx1250) — compile-verified
//
#include <hip/hip_runtime.h>
#include <math.h>

// ---------------------------------------------------------------------------
// AttentiveFP forward for MI455X (gfx1250, wave32, WMMA).
// Heavy GEMMs use V_WMMA_F32_16X16X4_F32 (full fp32 parity with reference).
// ---------------------------------------------------------------------------

#define BATCH 256
#define LATOM 128
#define KNBR  6
#define NFA   39
#define NFB   10
#define DIM   128
#define NROW  (BATCH * LATOM)   // 32768
#define NEGV  (-9e8f)

typedef float v2f __attribute__((ext_vector_type(2)));
typedef float v8f __attribute__((ext_vector_type(8)));

__device__ __forceinline__ float lrelu_f(float x) { return x > 0.f ? x : 0.01f * x; }
__device__ __forceinline__ float sigm_f(float x)  { return 1.f / (1.f + __expf(-x)); }
__device__ __forceinline__ float elu_f(float x)   { return x > 0.f ? x : (__expf(x) - 1.f); }

__device__ __forceinline__ float wredsum(float v) {
#pragma unroll
  for (int o = 16; o > 0; o >>= 1) v += __shfl_xor(v, o, 32);
  return v;  // butterfly: all 32 lanes hold the total
}

#define WMMA_F32(c, a, b) \
  __builtin_amdgcn_wmma_f32_16x16x4_f32(false, (a), false, (b), (short)0, (c), false, false)

// ---------------------------------------------------------------------------
// K1: atom_feature = leaky_relu(atom_list[32768,39] @ W[39,128] + b)
// block = 256 thr (8 waves) = 16 rows x 128 cols; one 16x16 tile per wave.
// ---------------------------------------------------------------------------
__global__ void atom_fc_kernel(const float* __restrict__ A, const float* __restrict__ W,
                               const float* __restrict__ bias, float* __restrict__ Out) {
  __shared__ float As[16 * 40];  // K padded 39 -> 40
  const int tid = threadIdx.x;
  const int row0 = blockIdx.x * 16;
  for (int e = tid; e < 16 * 40; e += 256) {
    int r = e / 40, j = e - r * 40;
    As[e] = (j < NFA) ? A[(size_t)(row0 + r) * NFA + j] : 0.f;
  }
  __syncthreads();
  const int wave = tid >> 5, lane = tid & 31;
  const int half = lane >> 4, r15 = lane & 15;
  const int col = wave * 16 + r15;
  v8f c = {0.f, 0.f, 0.f, 0.f, 0.f, 0.f, 0.f, 0.f};
#pragma unroll
  for (int kk = 0; kk < 10; kk++) {
    const int j0 = kk * 4 + half * 2;
    v2f a, b;
    a.x = As[r15 * 40 + j0];
    a.y = As[r15 * 40 + j0 + 1];
    b.x = (j0 < NFA) ? W[j0 * DIM + col] : 0.f;
    b.y = (j0 + 1 < NFA) ? W[(j0 + 1) * DIM + col] : 0.f;
    c = WMMA_F32(c, a, b);
  }
  const float bc = bias[col];
#pragma unroll
  for (int r = 0; r < 8; r++) {
    const int grow = row0 + r + half * 8;
    Out[(size_t)grow * DIM + col] = lrelu_f(c[r] + bc);
  }
}

// ---------------------------------------------------------------------------
// K2: fused neighbor gather (+49->128 FC for layer 0), align scores,
// masked softmax over K=6, weighted neighbor sum s[b,l,:] and sum-of-weights.
// One wave per (b,l); lane handles 4 features.
// ---------------------------------------------------------------------------
__global__ void nbr_attn_kernel(const float* __restrict__ cur, const float* __restrict__ gsrc,
                                const float* __restrict__ atom_list,
                                const float* __restrict__ bond_list,
                                const int* __restrict__ adeg, const int* __restrict__ bdeg,
                                const float* __restrict__ nbr_w, const float* __restrict__ nbr_b,
                                const float* __restrict__ aw, const float* __restrict__ ab_,
                                float* __restrict__ sbuf, float* __restrict__ sumw_out,
                                int useFC) {
  const int tid = threadIdx.x;
  const int wave = tid >> 5, lane = tid & 31;
  const int row = blockIdx.x * 8 + wave;
  const int b = row >> 7;

  const float4 c4 = ((const float4*)(cur + (size_t)row * DIM))[lane];
  const float4 at = ((const float4*)aw)[lane];        // align_w[0:128]
  const float4 abt = ((const float4*)aw)[32 + lane];  // align_w[128:256]
  const float p = wredsum(c4.x * at.x + c4.y * at.y + c4.z * at.z + c4.w * at.w);
  const float ab = ab_[0];

  float4 nb[KNBR];
  float q[KNBR], mk[KNBR];
#pragma unroll
  for (int k = 0; k < KNBR; k++) {
    const int ia = adeg[row * KNBR + k];
    mk[k] = (ia != (LATOM - 1)) ? 1.f : 0.f;
    float4 v;
    if (useFC) {
      const int ib = bdeg[row * KNBR + k];
      const float* ar = atom_list + ((size_t)b * LATOM + ia) * NFA;
      const float* br = bond_list + ((size_t)b * LATOM + ib) * NFB;
      float4 acc = {0.f, 0.f, 0.f, 0.f};
      for (int j = 0; j < NFA; j++) {
        const float x = ar[j];
        const float4 w = ((const float4*)(nbr_w + (size_t)j * DIM))[lane];
        acc.x += x * w.x; acc.y += x * w.y; acc.z += x * w.z; acc.w += x * w.w;
      }
      for (int j = 0; j < NFB; j++) {
        const float x = br[j];
        const float4 w = ((const float4*)(nbr_w + (size_t)(NFA + j) * DIM))[lane];
        acc.x += x * w.x; acc.y += x * w.y; acc.z += x * w.z; acc.w += x * w.w;
      }
      const float4 bb = ((const float4*)nbr_b)[lane];
      v.x = lrelu_f(acc.x + bb.x); v.y = lrelu_f(acc.y + bb.y);
      v.z = lrelu_f(acc.z + bb.z); v.w = lrelu_f(acc.w + bb.w);
    } else {
      v = ((const float4*)(gsrc + ((size_t)b * LATOM + ia) * DIM))[lane];
    }
    nb[k] = v;
    q[k] = wredsum(v.x * abt.x + v.y * abt.y + v.z * abt.z + v.w * abt.w);
  }

  float sc[KNBR], m = -1e30f;
#pragma unroll
  for (int k = 0; k < KNBR; k++) {
    sc[k] = lrelu_f(p + q[k] + ab) + (mk[k] == 0.f ? NEGV : 0.f);
    m = fmaxf(m, sc[k]);
  }
  float den = 0.f, w[KNBR];
#pragma unroll
  for (int k = 0; k < KNBR; k++) { w[k] = __expf(sc[k] - m); den += w[k]; }
  const float inv = 1.f / den;
  float sw = 0.f;
  float4 s = {0.f, 0.f, 0.f, 0.f};
#pragma unroll
  for (int k = 0; k < KNBR; k++) {
    const float wk = w[k] * inv * mk[k];
    sw += wk;
    s.x += wk * nb[k].x; s.y += wk * nb[k].y; s.z += wk * nb[k].z; s.w += wk * nb[k].w;
  }
  ((float4*)(sbuf + (size_t)row * DIM))[lane] = s;
  if (lane == 0) sumw_out[row] = sw;
}

// ---------------------------------------------------------------------------
// K3: Out = elu(S[M,128] @ W[128,128] + sumw[row]*bias)   (attend transform)
// ---------------------------------------------------------------------------
__global__ void gemm_elu_kernel(const float* __restrict__ S, const float* __restrict__ sumw,
                                const float* __restrict__ W, const float* __restrict__ bias,
                                float* __restrict__ Out) {
  __shared__ float As[16 * DIM];
  const int tid = threadIdx.x;
  const int row0 = blockIdx.x * 16;
  {
    const float4* gs = (const float4*)(S + (size_t)row0 * DIM);
    float4* ls = (float4*)As;
    ls[tid] = gs[tid];
    ls[tid + 256] = gs[tid + 256];
  }
  __syncthreads();
  const int wave = tid >> 5, lane = tid & 31;
  const int half = lane >> 4, r15 = lane & 15;
  const int col = wave * 16 + r15;
  v8f c = {0.f, 0.f, 0.f, 0.f, 0.f, 0.f, 0.f, 0.f};
#pragma unroll
  for (int kk = 0; kk < 32; kk++) {
    const int j0 = kk * 4 + half * 2;
    v2f a, b;
    a.x = As[r15 * DIM + j0];
    a.y = As[r15 * DIM + j0 + 1];
    b.x = W[(size_t)j0 * DIM + col];
    b.y = W[(size_t)(j0 + 1) * DIM + col];
    c = WMMA_F32(c, a, b);
  }
  const float bc = bias[col];
#pragma unroll
  for (int r = 0; r < 8; r++) {
    const int grow = row0 + r + half * 8;
    Out[(size_t)grow * DIM + col] = elu_f(c[r] + sumw[grow] * bc);
  }
}

// ---------------------------------------------------------------------------
// K4: fused GRU cell. gi = X@wihT (24 tiles), gh = H@whhT (24 tiles),
// 8 waves x 6 tiles each -> LDS G[16][768], then elementwise gates.
// In-place safe per 16-row block.
// ---------------------------------------------------------------------------
__global__ void gru_kernel(const float* __restrict__ X, const float* __restrict__ Hin,
                           const float* __restrict__ wihT, const float* __restrict__ whhT,
                           const float* __restrict__ bih, const float* __restrict__ bhh,
                           float* __restrict__ Hout, float* __restrict__ Act) {
  __shared__ float Xs[16 * DIM];
  __shared__ float Hs[16 * DIM];
  __shared__ float G[16 * 768];
  const int tid = threadIdx.x;
  const int row0 = blockIdx.x * 16;
  {
    const float4* gx = (const float4*)(X + (size_t)row0 * DIM);
    const float4* gh = (const float4*)(Hin + (size_t)row0 * DIM);
    float4* lx = (float4*)Xs;
    float4* lh = (float4*)Hs;
    lx[tid] = gx[tid];       lx[tid + 256] = gx[tid + 256];
    lh[tid] = gh[tid];       lh[tid + 256] = gh[tid + 256];
  }
  __syncthreads();
  const int wave = tid >> 5, lane = tid & 31;
  const int half = lane >> 4, r15 = lane & 15;
#pragma unroll
  for (int i = 0; i < 6; i++) {
    const int tt = wave * 6 + i;          // 0..47, wave-uniform gi/gh split
    const int is_gi = (tt < 24);
    const int nt = is_gi ? tt : tt - 24;  // 0..23
    const int col = nt * 16 + r15;        // 0..383
    const float* WT = is_gi ? wihT : whhT;
    const float* Asrc = is_gi ? Xs : Hs;
    v8f c = {0.f, 0.f, 0.f, 0.f, 0.f, 0.f, 0.f, 0.f};
#pragma unroll
    for (int kk = 0; kk < 32; kk++) {
      const int j0 = kk * 4 + half * 2;
      v2f a, b;
      a.x = Asrc[r15 * DIM + j0];
      a.y = Asrc[r15 * DIM + j0 + 1];
      b.x = WT[(size_t)j0 * 384 + col];
      b.y = WT[(size_t)(j0 + 1) * 384 + col];
      c = WMMA_F32(c, a, b);
    }
    const int base = is_gi ? 0 : 384;
#pragma unroll
    for (int r = 0; r < 8; r++) G[(r + half * 8) * 768 + base + col] = c[r];
  }
  __syncthreads();
#pragma unroll
  for (int i = 0; i < 8; i++) {
    const int e = tid + i * 256;
    const int rr = e >> 7, f = e & 127;
    const float ir = G[rr * 768 + f]       + bih[f];
    const float iz = G[rr * 768 + 128 + f] + bih[128 + f];
    const float in = G[rr * 768 + 256 + f] + bih[256 + f];
    const float hr = G[rr * 768 + 384 + f] + bhh[f];
    const float hz = G[rr * 768 + 512 + f] + bhh[128 + f];
    const float hn = G[rr * 768 + 640 + f] + bhh[256 + f];
    const float r_ = sigm_f(ir + hr);
    const float z  = sigm_f(iz + hz);
    const float n  = tanhf(in + r_ * hn);
    const float hp = Hs[rr * DIM + f];
    const float hv = (1.f - z) * n + z * hp;
    Hout[(size_t)(row0 + rr) * DIM + f] = hv;
    Act[(size_t)(row0 + rr) * DIM + f]  = fmaxf(hv, 0.f);
  }
}

// ---------------------------------------------------------------------------
// K5: masked pooling over atoms -> mol_feature, act_mol = relu(mol_feature)
// ---------------------------------------------------------------------------
__global__ void pool_kernel(const float* __restrict__ Act, const float* __restrict__ mask,
                            float* __restrict__ molF, float* __restrict__ actMol) {
  const int b = blockIdx.x, f = threadIdx.x;
  float acc = 0.f;
  for (int l = 0; l < LATOM; l++)
    acc += Act[((size_t)b * LATOM + l) * DIM + f] * mask[b * LATOM + l];
  molF[b * DIM + f] = acc;
  actMol[b * DIM + f] = fmaxf(acc, 0.f);
}

// ---------------------------------------------------------------------------
// K6: mol attention scores over L=128 atoms, masked softmax, weighted act sum
// ---------------------------------------------------------------------------
__global__ void mol_attn_kernel(const float* __restrict__ Act, const float* __restrict__ actMol,
                                const float* __restrict__ mask, const float* __restrict__ maw,
                                const float* __restrict__ mab_, float* __restrict__ molS,
                                float* __restrict__ molSumw) {
  __shared__ float sc[LATOM];
  __shared__ float tmp[LATOM];
  const int b = blockIdx.x, t = threadIdx.x;

  tmp[t] = actMol[b * DIM + t] * maw[t];
  __syncthreads();
  for (int o = 64; o > 0; o >>= 1) { if (t < o) tmp[t] += tmp[t + o]; __syncthreads(); }
  const float p = tmp[0];
  __syncthreads();

  const float* ar = Act + ((size_t)b * LATOM + t) * DIM;
  float qa = 0.f;
  for (int f = 0; f < DIM; f++) qa += ar[f] * maw[DIM + f];
  const float mk = mask[b * LATOM + t];
  const float s = lrelu_f(p + qa + mab_[0]) + (mk == 0.f ? NEGV : 0.f);
  tmp[t] = s;
  __syncthreads();
  for (int o = 64; o > 0; o >>= 1) { if (t < o) tmp[t] = fmaxf(tmp[t], tmp[t + o]); __syncthreads(); }
  const float mx = tmp[0];
  __syncthreads();
  const float e = __expf(s - mx);
  tmp[t] = e;
  __syncthreads();
  for (int o = 64; o > 0; o >>= 1) { if (t < o) tmp[t] += tmp[t + o]; __syncthreads(); }
  const float den = tmp[0];
  __syncthreads();
  const float w = e / den * mk;
  sc[t] = w;
  tmp[t] = w;
  __syncthreads();
  for (int o = 64; o > 0; o >>= 1) { if (t < o) tmp[t] += tmp[t + o]; __syncthreads(); }
  if (t == 0) molSumw[b] = tmp[0];

  float acc = 0.f;  // feature f = t
  for (int l = 0; l < LATOM; l++) acc += sc[l] * Act[((size_t)b * LATOM + l) * DIM + t];
  molS[b * DIM + t] = acc;
}

// ---------------------------------------------------------------------------
// K7: mol_prediction = mol_feature @ out_w + out_b
// ---------------------------------------------------------------------------
__global__ void pred_kernel(const float* __restrict__ molF, const float* __restrict__ ow,
                            const float* __restrict__ ob, float* __restrict__ out) {
  __shared__ float tmp[DIM];
  const int b = blockIdx.x, t = threadIdx.x;
  tmp[t] = molF[b * DIM + t] * ow[t];
  __syncthreads();
  for (int o = 64; o > 0; o >>= 1) { if (t < o) tmp[t] += tmp[t + o]; __syncthreads(); }
  if (t == 0) out[b] = tmp[0] + ob[0];
}

// K0: transpose [384,128] GRU weight -> [128,384] for coalesced WMMA B loads
__global__ void transpose_wT(const float* __restrict__ src, float* __restrict__ dst) {
  const int e = blockIdx.x * 256 + threadIdx.x;
  if (e < 384 * 128) { const int f = e / 128, j = e - f * 128; dst[j * 384 + f] = src[e]; }
}

// ---------------------------------------------------------------------------
extern "C" void kernel_launch(void* const* d_in, const int* in_sizes, int n_in,
                              void* d_out, int out_size, void* d_ws, size_t ws_size,
                              hipStream_t stream) {
  const float* atom_list   = (const float*)d_in[0];
  const float* bond_list   = (const float*)d_in[1];
  const float* atom_mask   = (const float*)d_in[2];
  const float* atom_fc_w   = (const float*)d_in[3];
  const float* atom_fc_b   = (const float*)d_in[4];
  const float* nbr_fc_w    = (const float*)d_in[5];
  const float* nbr_fc_b    = (const float*)d_in[6];
  const float* align_w     = (const float*)d_in[7];
  const float* align_b     = (const float*)d_in[8];
  const float* attend_w    = (const float*)d_in[9];
  const float* attend_b    = (const float*)d_in[10];
  const float* gru_wih     = (const float*)d_in[11];
  const float* gru_whh     = (const float*)d_in[12];
  const float* gru_bih     = (const float*)d_in[13];
  const float* gru_bhh     = (const float*)d_in[14];
  const float* mol_align_w = (const float*)d_in[15];
  const float* mol_align_b = (const float*)d_in[16];
  const float* mol_att_w   = (const float*)d_in[17];
  const float* mol_att_b   = (const float*)d_in[18];
  const float* mol_gru_wih = (const float*)d_in[19];
  const float* mol_gru_whh = (const float*)d_in[20];
  const float* mol_gru_bih = (const float*)d_in[21];
  const float* mol_gru_bhh = (const float*)d_in[22];
  const float* out_w       = (const float*)d_in[23];
  const float* out_b       = (const float*)d_in[24];
  const int*   adeg        = (const int*)d_in[25];
  const int*   bdeg        = (const int*)d_in[26];

  float* hfinal = (float*)d_out;                      // [B*L, D]
  float* pred   = hfinal + (size_t)NROW * DIM;        // [B]

  float* ws = (float*)d_ws;
  size_t o = 0;
  float* hbuf   = ws + o; o += (size_t)NROW * DIM;
  float* act    = ws + o; o += (size_t)NROW * DIM;
  float* sbuf   = ws + o; o += (size_t)NROW * DIM;
  float* ctx    = ws + o; o += (size_t)NROW * DIM;
  float* sumw   = ws + o; o += NROW;
  float* wT     = ws + o; o += (size_t)8 * 384 * 128;  // 0-2 wihT, 3-5 whhT, 6-7 mol
  float* molF0  = ws + o; o += BATCH * DIM;
  float* molF1  = ws + o; o += BATCH * DIM;
  float* actMol = ws + o; o += BATCH * DIM;
  float* molS   = ws + o; o += BATCH * DIM;
  float* molCtx = ws + o; o += BATCH * DIM;
  float* molSw  = ws + o; o += BATCH;

  const int WMAT = 384 * 128;  // 49152

  // transpose GRU weights (once per launch; deterministic)
  for (int d = 0; d < 3; d++) {
    transpose_wT<<<192, 256, 0, stream>>>(gru_wih + d * WMAT, wT + d * WMAT);
    transpose_wT<<<192, 256, 0, stream>>>(gru_whh + d * WMAT, wT + (3 + d) * WMAT);
  }
  transpose_wT<<<192, 256, 0, stream>>>(mol_gru_wih, wT + 6 * WMAT);
  transpose_wT<<<192, 256, 0, stream>>>(mol_gru_whh, wT + 7 * WMAT);

  // atom FC -> h (= atom_feature)
  atom_fc_kernel<<<NROW / 16, 256, 0, stream>>>(atom_list, atom_fc_w, atom_fc_b, hbuf);

  // layer 0 (uses nbr FC + bond features)
  nbr_attn_kernel<<<NROW / 8, 256, 0, stream>>>(hbuf, act, atom_list, bond_list, adeg, bdeg,
                                                nbr_fc_w, nbr_fc_b, align_w, align_b,
                                                sbuf, sumw, 1);
  gemm_elu_kernel<<<NROW / 16, 256, 0, stream>>>(sbuf, sumw, attend_w, attend_b, ctx);
  gru_kernel<<<NROW / 16, 256, 0, stream>>>(ctx, hbuf, wT, wT + 3 * WMAT,
                                            gru_bih, gru_bhh, hbuf, act);

  // layers 1..2 (neighbor features gathered from act)
  for (int d = 1; d < 3; d++) {
    nbr_attn_kernel<<<NROW / 8, 256, 0, stream>>>(act, act, atom_list, bond_list, adeg, bdeg,
                                                  nbr_fc_w, nbr_fc_b,
                                                  align_w + d * 256, align_b + d,
                                                  sbuf, sumw, 0);
    gemm_elu_kernel<<<NROW / 16, 256, 0, stream>>>(sbuf, sumw, attend_w + d * DIM * DIM,
                                                   attend_b + d * DIM, ctx);
    float* hout = (d == 2) ? hfinal : hbuf;  // final h goes straight to d_out
    gru_kernel<<<NROW / 16, 256, 0, stream>>>(ctx, hbuf, wT + d * WMAT, wT + (3 + d) * WMAT,
                                              gru_bih + d * 384, gru_bhh + d * 384, hout, act);
  }

  // molecule readout
  pool_kernel<<<BATCH, 128, 0, stream>>>(act, atom_mask, molF0, actMol);
  float* mf = molF0;
  float* mfn = molF1;
  for (int t = 0; t < 2; t++) {
    mol_attn_kernel<<<BATCH, 128, 0, stream>>>(act, actMol, atom_mask, mol_align_w,
                                               mol_align_b, molS, molSw);
    gemm_elu_kernel<<<BATCH / 16, 256, 0, stream>>>(molS, molSw, mol_att_w, mol_att_b, molCtx);
    gru_kernel<<<BATCH / 16, 256, 0, stream>>>(molCtx, mf, wT + 6 * WMAT, wT + 7 * WMAT,
                                               mol_gru_bih, mol_gru_bhh, mfn, actMol);
    float* t2 = mf; mf = mfn; mfn = t2;
  }
  pred_kernel<<<BATCH, 128, 0, stream>>>(mf, out_w, out_b, pred);
}